// VectorQuantizer_28759101014238
// MI455X (gfx1250) — compile-verified
//
#include <hip/hip_runtime.h>

typedef __attribute__((ext_vector_type(2))) float v2f;
typedef __attribute__((ext_vector_type(8))) float v8f;

// Types for the async global->LDS builtin (from hipcc diagnostic: param 0 is
// 'int __vector_size__(16) addrspace(1)*').
typedef __attribute__((__vector_size__(16))) int async_v4i;
typedef __attribute__((address_space(1))) async_v4i* gv4i_p;
typedef __attribute__((address_space(3))) async_v4i* lv4i_p;

#define VOCAB 1024
#define EMBED 64
#define NVEC 32768          // 32*32*32 vectors
#define OUT_ELEMS 2097152   // 32*64*32*32
#define LDS_PITCH 68        // floats per codebook row in LDS (bank-conflict pad)
#define LDS_BYTES (VOCAB * LDS_PITCH * 4)  // 278528 B (< 320KB WGP LDS)

// ---------------------------------------------------------------------------
// Kernel 0: codebook row norms + zero histogram
// ---------------------------------------------------------------------------
__global__ void vq_prep(const float* __restrict__ cb,
                        float* __restrict__ cnorm,
                        int* __restrict__ hist) {
  int v = blockIdx.x * blockDim.x + threadIdx.x;
  if (v < VOCAB) {
    const float4* row = (const float4*)(cb + v * EMBED);
    float s = 0.0f;
#pragma unroll
    for (int i = 0; i < 16; ++i) {
      float4 c = row[i];
      s += c.x * c.x + c.y * c.y + c.z * c.z + c.w * c.w;
    }
    cnorm[v] = s;
    hist[v] = 0;
  }
}

// ---------------------------------------------------------------------------
// Kernel 1: WMMA distance matmul + argmin.
// 256 threads = 8 waves per block; each wave owns one 16-row tile.
// Codebook staged once per block into padded LDS via async global->LDS,
// then B fragments come from ds_load_b64 (conflict-free via pitch 68).
// dist = ||c||^2 - 2 * (x . c)   (||x||^2 constant per row, dropped)
// ---------------------------------------------------------------------------
__global__ void __launch_bounds__(256)
vq_argmin(const float* __restrict__ x,      // [32,64,32,32] channels-first
          const float* __restrict__ cb,     // [1024,64] row-major
          const float* __restrict__ cnorm,  // [1024]
          int* __restrict__ idx_out) {      // [NVEC]
  extern __shared__ float lds_cb[];         // [VOCAB * LDS_PITCH]

  // ---- Stage codebook into LDS (padded pitch), async DMA if available ----
  // 16384 x 16B chunks; 256 threads -> 64 chunks each.
  {
    const int tid = threadIdx.x;
#if __has_builtin(__builtin_amdgcn_global_load_async_to_lds_b128)
#pragma unroll 4
    for (int i = 0; i < 64; ++i) {
      const int ch = tid * 64 + i;        // chunk id
      const int row = ch >> 4;
      const int cg = ch & 15;             // 16B-group within row
      __builtin_amdgcn_global_load_async_to_lds_b128(
          (gv4i_p)(cb + row * EMBED + cg * 4),
          (lv4i_p)(lds_cb + row * LDS_PITCH + cg * 4),
          /*offset=*/0, /*cpol=*/0);
    }
#if __has_builtin(__builtin_amdgcn_s_wait_asynccnt)
    __builtin_amdgcn_s_wait_asynccnt(0);
#endif
#else
#pragma unroll 4
    for (int i = 0; i < 64; ++i) {
      const int ch = tid * 64 + i;
      const int row = ch >> 4;
      const int cg = ch & 15;
      float4 v = ((const float4*)(cb + row * EMBED))[cg];
      *(float4*)(lds_cb + row * LDS_PITCH + cg * 4) = v;
    }
#endif
    __syncthreads();
  }

  const int lane = threadIdx.x & 31;
  const int wave = threadIdx.x >> 5;        // 0..7
  const int half = lane >> 4;               // 0 or 1
  const int lq = lane & 15;
  const int m0 = (blockIdx.x * 8 + wave) << 4;  // first row of this wave's tile

  // --- A fragments: row n = m0+lq, flat[n,d] = x[(n>>10)*65536 + d*1024 + (n&1023)]
  const int n = m0 + lq;
  const float* abase = x + (n >> 10) * 65536 + (n & 1023);
  v2f a[16];
#pragma unroll
  for (int kc = 0; kc < 16; ++kc) {
    const int d = kc * 4 + 2 * half;  // K=0,1 on lanes 0-15 ; K=2,3 on lanes 16-31
    a[kc].x = abase[d * 1024];
    a[kc].y = abase[(d + 1) * 1024];
  }

  float bestD[8];
  int bestI[8];
#pragma unroll
  for (int j = 0; j < 8; ++j) { bestD[j] = 3.4e38f; bestI[j] = 0; }

  const v2f* lcb2 = (const v2f*)lds_cb;     // LDS_PITCH is even
  for (int ct = 0; ct < 64; ++ct) {
    const int code = (ct << 4) + lq;        // this lane's codebook column
    v8f acc = {};
#pragma unroll
    for (int kc = 0; kc < 16; ++kc) {
      // B fragment from LDS: float2 at (code*LDS_PITCH + 2*half + kc*4)
      v2f b = lcb2[code * (LDS_PITCH / 2) + half + kc * 2];
      acc = __builtin_amdgcn_wmma_f32_16x16x4_f32(
          /*neg_a=*/false, a[kc], /*neg_b=*/false, b,
          /*c_mod=*/(short)0, acc, /*reuse_a=*/false, /*reuse_b=*/false);
    }
    const float cn = cnorm[code];
#pragma unroll
    for (int j = 0; j < 8; ++j) {
      float dist = cn - 2.0f * acc[j];      // row m0 + j + 8*half, col `code`
      if (dist < bestD[j]) { bestD[j] = dist; bestI[j] = code; }
    }
  }

  // Reduce across the 16 lanes of each half (masks 8..1 stay within a half).
#pragma unroll
  for (int j = 0; j < 8; ++j) {
#pragma unroll
    for (int m = 8; m >= 1; m >>= 1) {
      float d2 = __shfl_xor(bestD[j], m, 32);
      int i2 = __shfl_xor(bestI[j], m, 32);
      if (d2 < bestD[j] || (d2 == bestD[j] && i2 < bestI[j])) {
        bestD[j] = d2; bestI[j] = i2;
      }
    }
    if (lq == 0) idx_out[m0 + j + 8 * half] = bestI[j];
  }
}

// ---------------------------------------------------------------------------
// Kernel 2: quantized output (straight-through == gather) + loss partials.
// Each thread handles 4 consecutive elements of the [B,C,H,W] output.
// ---------------------------------------------------------------------------
__global__ void __launch_bounds__(256)
vq_out_loss(const float* __restrict__ x, const float* __restrict__ cb,
            const int* __restrict__ idx, float* __restrict__ out,
            float* __restrict__ partial) {
  __shared__ float sred[256];
  const int t4 = blockIdx.x * 256 + threadIdx.x;
  const int t = t4 * 4;
  const int c = (t >> 10) & 63;
  const int n0 = ((t >> 16) << 10) | (t & 1023);  // 4 consecutive n
  float4 xin = ((const float4*)x)[t4];
  float q0 = cb[idx[n0 + 0] * 64 + c];
  float q1 = cb[idx[n0 + 1] * 64 + c];
  float q2 = cb[idx[n0 + 2] * 64 + c];
  float q3 = cb[idx[n0 + 3] * 64 + c];
  // out region is only 4-byte aligned (d_out+1) -> scalar stores
  out[t + 0] = q0; out[t + 1] = q1; out[t + 2] = q2; out[t + 3] = q3;
  float d0 = q0 - xin.x, d1 = q1 - xin.y, d2 = q2 - xin.z, d3 = q3 - xin.w;
  sred[threadIdx.x] = d0 * d0 + d1 * d1 + d2 * d2 + d3 * d3;
  __syncthreads();
  for (int off = 128; off > 0; off >>= 1) {
    if (threadIdx.x < off) sred[threadIdx.x] += sred[threadIdx.x + off];
    __syncthreads();
  }
  if (threadIdx.x == 0) partial[blockIdx.x] = sred[0];
}

// ---------------------------------------------------------------------------
// Kernel 3: one-hot encodings [NVEC, 1024]. Region is 8-byte aligned -> float2.
// Each thread writes 8 floats (4 float2) of one row.
// ---------------------------------------------------------------------------
__global__ void __launch_bounds__(256)
vq_enc(const int* __restrict__ idx, float2* __restrict__ enc) {
  const int t = blockIdx.x * 256 + threadIdx.x;
  const int base = t * 4;              // float2 index
  const int n = base >> 9;             // 512 float2 per row
  const int code = idx[n];
  const int col0 = (base & 511) * 2;   // starting float column
#pragma unroll
  for (int i = 0; i < 4; ++i) {
    const int col = col0 + 2 * i;
    float2 v;
    v.x = (code == col) ? 1.0f : 0.0f;
    v.y = (code == col + 1) ? 1.0f : 0.0f;
    enc[base + i] = v;
  }
}

// ---------------------------------------------------------------------------
// Kernel 4: histogram of code usage (integer atomics -> deterministic result)
// ---------------------------------------------------------------------------
__global__ void __launch_bounds__(256)
vq_hist(const int* __restrict__ idx, int* __restrict__ hist) {
  const int n = blockIdx.x * 256 + threadIdx.x;
  atomicAdd(&hist[idx[n]], 1);
}

// ---------------------------------------------------------------------------
// Kernel 5: finalize loss + perplexity. Single block of 1024 threads.
// ---------------------------------------------------------------------------
__global__ void __launch_bounds__(1024)
vq_final(const int* __restrict__ hist, const float* __restrict__ partial,
         float* __restrict__ loss_out, float* __restrict__ ppl_out) {
  __shared__ float sred[1024];
  const int v = threadIdx.x;
  const float p = (float)hist[v] * (1.0f / (float)NVEC);
  sred[v] = p * logf(p + 1e-10f);
  __syncthreads();
  for (int off = 512; off > 0; off >>= 1) {
    if (v < off) sred[v] += sred[v + off];
    __syncthreads();
  }
  if (v == 0) {
    *ppl_out = expf(-sred[0]);
    float s = 0.0f;
    for (int i = 0; i < 2048; ++i) s += partial[i];  // fixed order: deterministic
    *loss_out = 1.25f * s * (1.0f / (float)OUT_ELEMS);
  }
}

// ---------------------------------------------------------------------------
extern "C" void kernel_launch(void* const* d_in, const int* in_sizes, int n_in,
                              void* d_out, int out_size, void* d_ws, size_t ws_size,
                              hipStream_t stream) {
  (void)in_sizes; (void)n_in; (void)out_size; (void)ws_size;
  const float* x = (const float*)d_in[0];   // [32,64,32,32] f32
  const float* cb = (const float*)d_in[1];  // [1024,64] f32

  float* out = (float*)d_out;
  float* loss = out;                        // [0]
  float* qout = out + 1;                    // [1 .. 1+OUT_ELEMS)
  float* ppl = out + 1 + OUT_ELEMS;         // [1+OUT_ELEMS]
  float* enc = out + 2 + OUT_ELEMS;         // [2+OUT_ELEMS ..), 8B aligned

  char* ws = (char*)d_ws;
  int* idx = (int*)ws;                                   // NVEC ints
  float* cnorm = (float*)(ws + NVEC * 4);                // VOCAB floats
  int* hist = (int*)(ws + NVEC * 4 + VOCAB * 4);         // VOCAB ints
  float* partial = (float*)(ws + NVEC * 4 + VOCAB * 8);  // 2048 floats

  vq_prep<<<4, 256, 0, stream>>>(cb, cnorm, hist);
  vq_argmin<<<NVEC / (16 * 8), 256, LDS_BYTES, stream>>>(x, cb, cnorm, idx);
  vq_out_loss<<<OUT_ELEMS / 1024, 256, 0, stream>>>(x, cb, idx, qout, partial);
  vq_enc<<<(NVEC * VOCAB / 8) / 256, 256, 0, stream>>>(idx, (float2*)enc);
  vq_hist<<<NVEC / 256, 256, 0, stream>>>(idx, hist);
  vq_final<<<1, 1024, 0, stream>>>(hist, partial, loss, ppl);
}